// BackProjNet_21225728377452
// MI455X (gfx1250) — compile-verified
//
#include <hip/hip_runtime.h>
#include <stdint.h>

// ---- CT geometry (must match reference) ----
#define CH        8
#define NRAY      92160     // VIEWS*NDETU = 180*512
#define NVOX      65536     // 256*256
#define JTOT      360       // VIEWS*EXTENT = 180*2
#define CHUNK     128       // rays per async chunk (32 lanes x 4 rays)
#define NCHUNK    3         // ceil(360/128)
#define WPB       8         // waves per block (256 threads)
#define SCALE_F   0.008726646259971648f  // 2*pi / (2*180*2)

// Low 32 bits of a generic pointer to __shared__ == LDS byte offset.
__device__ __forceinline__ uint32_t lds_off(const void* p) {
    return (uint32_t)(uint64_t)(uintptr_t)p;
}

// CDNA5 async DMA: 16B/lane memory -> LDS, tracked by ASYNCcnt.
// NT hint: idx/weight stream through once; keep L2 for the gathered xT.
__device__ __forceinline__ void async_ld_b128(uint32_t ldsOff, const void* g) {
    asm volatile("global_load_async_to_lds_b128 %0, %1, off th:TH_LOAD_NT"
                 :: "v"(ldsOff), "v"((uint64_t)(uintptr_t)g)
                 : "memory");
}

// ---- prep: x[8][92160] -> xT[92160][8] so each ray's channels are 32B ----
__global__ __launch_bounds__(256) void transpose_x_kernel(
    const float* __restrict__ x, float* __restrict__ xT) {
    int t = blockIdx.x * blockDim.x + threadIdx.x;
    if (t < NRAY * CH) {
        int r = t >> 3;
        int c = t & 7;
        xT[t] = x[c * NRAY + r];   // coalesced store
    }
}

// ---- main: one wave32 per voxel; lane l owns rays 4l..4l+3 per chunk ----
template <bool USE_XT>
__global__ __launch_bounds__(256) void backproj_kernel(
    const float*  __restrict__ x,        // [8][92160]
    const float*  __restrict__ weight,   // [NVOX*JTOT]
    const float*  __restrict__ bias,     // [NVOX]
    const int*    __restrict__ indices,  // [NVOX*JTOT]
    const float4* __restrict__ xT,       // [92160][2] float4 (ws)
    float*        __restrict__ out)      // [8][NVOX]
{
    alignas(16) __shared__ uint4  sIdx[WPB][NCHUNK][32];
    alignas(16) __shared__ float4 sW  [WPB][NCHUNK][32];

    const int lane  = threadIdx.x & 31;
    const int wv    = threadIdx.x >> 5;
    const int voxel = blockIdx.x * WPB + wv;       // grid = NVOX/WPB exactly
    const long nIdxM4 = (long)NVOX * JTOT - 4;     // clamp so r..r+3 in-bounds

    float a0=0.f,a1=0.f,a2=0.f,a3=0.f,a4=0.f,a5=0.f,a6=0.f,a7=0.f;

    // --- preload all 3 chunks of (idx, w) into this wave's LDS slots ---
#pragma unroll
    for (int k = 0; k < NCHUNK; ++k) {
        long r = (long)voxel * JTOT + k * CHUNK + 4 * lane;
        r = (r > nIdxM4) ? nIdxM4 : r;             // 16B-aligned clamp
        async_ld_b128(lds_off(&sIdx[wv][k][lane]), indices + r);
        async_ld_b128(lds_off(&sW  [wv][k][lane]), weight  + r);
    }

#pragma unroll
    for (int k = 0; k < NCHUNK; ++k) {
        // Async loads complete in order: chunk k done when <= 2*(NCHUNK-1-k)
        // of the 2-per-chunk loads remain outstanding.
        if (k == 0)      asm volatile("s_wait_asynccnt 4" ::: "memory");
        else if (k == 1) asm volatile("s_wait_asynccnt 2" ::: "memory");
        else             asm volatile("s_wait_asynccnt 0" ::: "memory");

        const uint4  iv = sIdx[wv][k][lane];       // ds_load_b128
        const float4 wf = sW  [wv][k][lane];       // ds_load_b128
        const int j0 = k * CHUNK + 4 * lane;       // first ray of this lane

        const uint32_t ids[4] = {iv.x, iv.y, iv.z, iv.w};
        const float    ws [4] = {wf.x, wf.y, wf.z, wf.w};

#pragma unroll
        for (int s = 0; s < 4; ++s) {
            float wz = ws[s];
            if (j0 + s >= JTOT) wz = 0.f;          // tail rays contribute 0
            const uint32_t idxv = ids[s];

            if (USE_XT) {
                // 32B contiguous gather of all 8 channels (L2-resident xT)
                const float4 lo = xT[(size_t)idxv * 2 + 0];
                const float4 hi = xT[(size_t)idxv * 2 + 1];
                a0 = fmaf(wz, lo.x, a0);  a1 = fmaf(wz, lo.y, a1);
                a2 = fmaf(wz, lo.z, a2);  a3 = fmaf(wz, lo.w, a3);
                a4 = fmaf(wz, hi.x, a4);  a5 = fmaf(wz, hi.y, a5);
                a6 = fmaf(wz, hi.z, a6);  a7 = fmaf(wz, hi.w, a7);
            } else {
                a0 = fmaf(wz, x[0*NRAY + idxv], a0);
                a1 = fmaf(wz, x[1*NRAY + idxv], a1);
                a2 = fmaf(wz, x[2*NRAY + idxv], a2);
                a3 = fmaf(wz, x[3*NRAY + idxv], a3);
                a4 = fmaf(wz, x[4*NRAY + idxv], a4);
                a5 = fmaf(wz, x[5*NRAY + idxv], a5);
                a6 = fmaf(wz, x[6*NRAY + idxv], a6);
                a7 = fmaf(wz, x[7*NRAY + idxv], a7);
            }
        }
    }

    // --- 32-lane butterfly reduction of the 8 channel partials ---
#pragma unroll
    for (int m = 16; m >= 1; m >>= 1) {
        a0 += __shfl_xor(a0, m, 32);  a1 += __shfl_xor(a1, m, 32);
        a2 += __shfl_xor(a2, m, 32);  a3 += __shfl_xor(a3, m, 32);
        a4 += __shfl_xor(a4, m, 32);  a5 += __shfl_xor(a5, m, 32);
        a6 += __shfl_xor(a6, m, 32);  a7 += __shfl_xor(a7, m, 32);
    }

    // All lanes now hold all 8 sums; lanes 0..7 store one channel each.
    float val = a0;
    val = (lane == 1) ? a1 : val;
    val = (lane == 2) ? a2 : val;
    val = (lane == 3) ? a3 : val;
    val = (lane == 4) ? a4 : val;
    val = (lane == 5) ? a5 : val;
    val = (lane == 6) ? a6 : val;
    val = (lane == 7) ? a7 : val;
    if (lane < CH) {
        out[lane * NVOX + voxel] = val * SCALE_F + bias[voxel];
    }
}

extern "C" void kernel_launch(void* const* d_in, const int* in_sizes, int n_in,
                              void* d_out, int out_size, void* d_ws, size_t ws_size,
                              hipStream_t stream) {
    const float* x       = (const float*)d_in[0];  // [1,8,92160]
    const float* weight  = (const float*)d_in[1];  // [23592960]
    const float* bias    = (const float*)d_in[2];  // [65536]
    const int*   indices = (const int*)  d_in[3];  // [23592960]
    float*       out     = (float*)d_out;          // [1,8,256,256]

    const size_t xtBytes = (size_t)NRAY * CH * sizeof(float);  // ~2.95 MB
    if (ws_size >= xtBytes) {
        float* xT = (float*)d_ws;
        const int n = NRAY * CH;
        transpose_x_kernel<<<(n + 255) / 256, 256, 0, stream>>>(x, xT);
        backproj_kernel<true><<<NVOX / WPB, 256, 0, stream>>>(
            x, weight, bias, indices, (const float4*)d_ws, out);
    } else {
        backproj_kernel<false><<<NVOX / WPB, 256, 0, stream>>>(
            x, weight, bias, indices, nullptr, out);
    }
}